// Correlation1d_74363063763057
// MI455X (gfx1250) — compile-verified
//
#include <hip/hip_runtime.h>

typedef float v2f __attribute__((ext_vector_type(2)));
typedef float v8f __attribute__((ext_vector_type(8)));

// Problem constants (fixed by the reference).
constexpr int Bn = 8, Cc = 128, Hh = 96, Ww = 320;
constexpr int HW = Hh * Ww;
constexpr int ND = 9;                       // displacement channels (-4..4)
constexpr int NQ = Cc / 4;                  // 32 K-chunks of 4 channels
constexpr int TILES_PER_SEG = 4;            // 4 x-tiles of 16 -> 64 cols per wave
constexpr int SEGS = Ww / (16 * TILES_PER_SEG);  // 5
constexpr int WAVES_PER_BLOCK = 8;

// Load one 16-column x2p block as 32 B fragments (4x16 K x N each).
// x2p[c,j] = x2[c, j-4] zero-padded outside [0, W).
// B layout (f32 4x16): lane n = lane&15, VGPR v / lane-group g -> K = v + 2g.
__device__ __forceinline__ void load_bblock(v2f* Bb, const float* __restrict__ x2row,
                                            int j0, int g, int nn) {
  int xsrc = j0 + nn - 4;
  bool valid = ((unsigned)xsrc < (unsigned)Ww);
#pragma unroll
  for (int q = 0; q < NQ; ++q) {
    int c = 4 * q + 2 * g;
    float bx = 0.0f, by = 0.0f;
    if (valid) {
      const float* p = x2row + c * HW + xsrc;
      bx = p[0];
      by = p[HW];
    }
    Bb[q].x = bx;
    Bb[q].y = by;
  }
}

// Load one A fragment chunk (16x4 M x K) for tile rows starting at x0.
// A layout (f32 16x4): lane m = lane&15, VGPR v / group g -> K = v + 2g.
// x1 is streamed exactly once per wave -> non-temporal.
__device__ __forceinline__ v2f load_afrag(const float* __restrict__ x1row, int x0,
                                          int q, int g, int mm) {
  int c = 4 * q + 2 * g;
  const float* p = x1row + c * HW + x0 + mm;
  v2f a;
  a.x = __builtin_nontemporal_load(p);
  a.y = __builtin_nontemporal_load(p + HW);
  return a;
}

__device__ __forceinline__ v8f wmma4(const v2f a, const v2f b, const v8f c) {
  return __builtin_amdgcn_wmma_f32_16x16x4_f32(false, a, false, b, (short)0, c,
                                               false, false);
}

// Scatter the 9-wide band of the two 16x16 Gram tiles (rows x0..x0+15,
// cols x0..x0+31) into out[b, t, y, x0+m], t = j - x.  Write-once streaming
// output -> non-temporal stores.
// C/D layout: lane n = lane&15, VGPR r / group g -> M = r + 8g.
__device__ __forceinline__ void store_tile(float* __restrict__ outbase, int x0,
                                           int g, int nn, const v8f d0,
                                           const v8f d1) {
  const float invC = 1.0f / (float)Cc;
#pragma unroll
  for (int r = 0; r < 8; ++r) {
    int mrow = r + 8 * g;
    int t0 = nn - mrow;  // column j = x0 + nn
    if ((unsigned)t0 <= 8u)
      __builtin_nontemporal_store(d0[r] * invC, outbase + t0 * HW + x0 + mrow);
    int t1 = nn + 16 - mrow;  // column j = x0 + 16 + nn
    if ((unsigned)t1 <= 8u)
      __builtin_nontemporal_store(d1[r] * invC, outbase + t1 * HW + x0 + mrow);
  }
}

__global__ __launch_bounds__(WAVES_PER_BLOCK * 32) void
Correlation1d_74363063763057_kernel(const float* __restrict__ x1,
                                    const float* __restrict__ x2,
                                    float* __restrict__ out) {
  int wid = blockIdx.x * WAVES_PER_BLOCK + ((int)threadIdx.x >> 5);
  int lane = (int)threadIdx.x & 31;
  int nn = lane & 15;  // N (col) for B/C/D frags, M (row) for A frags
  int g = lane >> 4;   // lane half

  int seg = wid % SEGS;
  int row = wid / SEGS;
  if (row >= Bn * Hh) return;  // wave-uniform guard: EXEC stays all-ones
  int y = row % Hh;
  int b = row / Hh;

  const float* x1row = x1 + (size_t)b * Cc * HW + (size_t)y * Ww;
  const float* x2row = x2 + (size_t)b * Cc * HW + (size_t)y * Ww;
  float* outbase = out + (size_t)b * ND * HW + (size_t)y * Ww;

  int xs = seg * (16 * TILES_PER_SEG);

  v8f zero = {};
  v8f d0a = zero, d0b = zero;  // even/odd-chunk partials, D0 of current tile
  v8f d1a = zero, d1b = zero;  // even/odd-chunk partials, D1 of previous tile
  v8f d0_done = zero;          // completed D0 of previous tile

  v2f Aa[NQ];  // cached A fragments of the current tile (reused next block)
  v2f Bb[NQ];  // sliding 16-column x2p block of B fragments

  // Block 0: columns j = xs..xs+15; builds tile 0's D0 and caches A(tile 0).
  load_bblock(Bb, x2row, xs, g, nn);
#pragma unroll
  for (int q = 0; q < NQ; ++q) {
    Aa[q] = load_afrag(x1row, xs, q, g, nn);
    if (q & 1)
      d0b = wmma4(Aa[q], Bb[q], d0b);
    else
      d0a = wmma4(Aa[q], Bb[q], d0a);
  }
  d0_done = d0a + d0b;

  // Middle blocks k = 1..3: block k finishes tile k-1 (D1, using cached A)
  // and builds tile k (D0, loading + caching its A fragments).
  for (int kblk = 1; kblk < TILES_PER_SEG; ++kblk) {
    int j0 = xs + 16 * kblk;
    load_bblock(Bb, x2row, j0, g, nn);
    d0a = zero;
    d0b = zero;
#pragma unroll
    for (int q = 0; q < NQ; ++q) {
      v2f an = load_afrag(x1row, j0, q, g, nn);
      if (q & 1) {
        d1b = wmma4(Aa[q], Bb[q], d1b);
        d0b = wmma4(an, Bb[q], d0b);
      } else {
        d1a = wmma4(Aa[q], Bb[q], d1a);
        d0a = wmma4(an, Bb[q], d0a);
      }
      Aa[q] = an;
    }
    store_tile(outbase, j0 - 16, g, nn, d0_done, d1a + d1b);
    d0_done = d0a + d0b;
    d1a = zero;
    d1b = zero;
  }

  // Final block: columns j = xs+64..xs+79; finishes the last tile's D1.
  {
    int j0 = xs + 16 * TILES_PER_SEG;
    load_bblock(Bb, x2row, j0, g, nn);
#pragma unroll
    for (int q = 0; q < NQ; ++q) {
      if (q & 1)
        d1b = wmma4(Aa[q], Bb[q], d1b);
      else
        d1a = wmma4(Aa[q], Bb[q], d1a);
    }
    store_tile(outbase, j0 - 16, g, nn, d0_done, d1a + d1b);
  }
}

extern "C" void kernel_launch(void* const* d_in, const int* in_sizes, int n_in,
                              void* d_out, int out_size, void* d_ws, size_t ws_size,
                              hipStream_t stream) {
  (void)in_sizes; (void)n_in; (void)out_size; (void)d_ws; (void)ws_size;
  const float* x1 = (const float*)d_in[0];
  const float* x2 = (const float*)d_in[1];
  float* out = (float*)d_out;

  int totalWaves = Bn * Hh * SEGS;  // 8*96*5 = 3840 waves
  int blocks = (totalWaves + WAVES_PER_BLOCK - 1) / WAVES_PER_BLOCK;  // 480
  Correlation1d_74363063763057_kernel<<<blocks, WAVES_PER_BLOCK * 32, 0, stream>>>(
      x1, x2, out);
}